// DYDConv2d_O_71476845740311
// MI455X (gfx1250) — compile-verified
//
#include <hip/hip_runtime.h>

typedef float v4f __attribute__((ext_vector_type(4)));

#define BC_TOTAL   512        // B*C = 8*64
#define HW         128        // H = W = outH = outW
#define PLANE      16384      // 128*128
#define TILE_ROWS  16         // output rows per workgroup
#define HALO_ROWS  18         // TILE_ROWS + 2
#define TS         136        // LDS row stride (dwords); interior starts at +4 -> 16B aligned
#define LDS_DW     (HALO_ROWS * TS)   // 2448 dwords = 9792 B
#define NTHREADS   256

__global__ __launch_bounds__(NTHREADS)
void DYDConv2d_O_71476845740311_kernel(const float* __restrict__ inp,
                                       const float* __restrict__ wgt,
                                       float* __restrict__ out)
{
    __shared__ __align__(16) float tile[LDS_DW];

    const int t   = threadIdx.x;
    const int bc  = blockIdx.x >> 3;                 // b*C + c  (0..511)
    const int oh0 = (blockIdx.x & 7) * TILE_ROWS;    // first output row of tile

    // ---- zero whole LDS tile: halo columns/rows must read as 0 (padding) ----
    for (int q = t; q < LDS_DW / 4; q += NTHREADS) {
        v4f z = {0.f, 0.f, 0.f, 0.f};
        *(v4f*)&tile[q * 4] = z;
    }
    __syncthreads();

    // ---- async-stage input rows [oh0-1 .. oh0+16] into LDS interior ----
    // tile[r_t*TS + 4 + w] <- input[bc][oh0-1+r_t][w]   (w = 0..127)
    const float* plane = inp + (size_t)bc * PLANE;
    #pragma unroll
    for (int k = 0; k < 3; ++k) {
        int l = t + k * NTHREADS;                    // 0..575 = 18 rows * 32 segs
        if (l < HALO_ROWS * 32) {
            int row  = l >> 5;                       // 0..17
            int seg  = l & 31;                       // 0..31 (x4 floats)
            int grow = oh0 - 1 + row;                // global input row
            if (0 <= grow && grow < HW) {
                unsigned lds_off = (unsigned)(size_t)&tile[row * TS + 4 + seg * 4];
                const float* gp  = plane + grow * HW + seg * 4;
                asm volatile("global_load_async_to_lds_b128 %0, %1, off"
                             :: "v"(lds_off), "v"(gp) : "memory");
            }
        }
    }
    asm volatile("s_wait_asynccnt 0x0" ::: "memory");
    __syncthreads();

    // ---- compute: 2 float4 output quads per thread ----
    const float* wq0  = wgt + (size_t)bc * 9 * PLANE;
    float*       outp = out + (size_t)bc * PLANE;

    #pragma unroll
    for (int k = 0; k < 2; ++k) {
        int qi   = t + k * NTHREADS;                 // 0..511
        int r    = qi >> 5;                          // 0..15  tile-local output row
        int cq   = (qi & 31) << 2;                   // 0..124 output col (x4)
        int orow = oh0 + r;
        const float* wq = wq0 + (size_t)orow * HW + cq;

        v4f acc = {0.f, 0.f, 0.f, 0.f};
        #pragma unroll
        for (int i = 0; i < 3; ++i) {
            // need tile cols (c_t) cq+3 .. cq+8 of halo row r+i
            int   base = (r + i) * TS + cq;
            float s0   = tile[base + 3];             // ds_load_b32
            v4f   sv   = *(const v4f*)&tile[base + 4]; // 16B-aligned ds_load_b128
            float s5   = tile[base + 8];             // ds_load_b32
            float s[6] = {s0, sv.x, sv.y, sv.z, sv.w, s5};
            #pragma unroll
            for (int j = 0; j < 3; ++j) {
                // per-tap weights contiguous over ow -> coalesced NT b128 stream
                v4f w4 = __builtin_nontemporal_load(
                    (const v4f*)(wq + (size_t)(i * 3 + j) * PLANE));
                acc.x += w4.x * s[j + 0];
                acc.y += w4.y * s[j + 1];
                acc.z += w4.z * s[j + 2];
                acc.w += w4.w * s[j + 3];
            }
        }
        __builtin_nontemporal_store(acc, (v4f*)(outp + (size_t)orow * HW + cq));
    }
}

extern "C" void kernel_launch(void* const* d_in, const int* in_sizes, int n_in,
                              void* d_out, int out_size, void* d_ws, size_t ws_size,
                              hipStream_t stream) {
    const float* inp = (const float*)d_in[0];   // (8,64,128,128) f32
    const float* wgt = (const float*)d_in[1];   // (8,64,3,3,128,128) f32
    float*       out = (float*)d_out;           // (8,64,128,128) f32

    dim3 grid(BC_TOTAL * (HW / TILE_ROWS));     // 512 * 8 = 4096 workgroups
    DYDConv2d_O_71476845740311_kernel<<<grid, NTHREADS, 0, stream>>>(inp, wgt, out);
}